// PhyDGalerkinConv_6897717477692
// MI455X (gfx1250) — compile-verified
//
#include <hip/hip_runtime.h>
#include <hip/hip_bf16.h>

typedef __attribute__((ext_vector_type(2))) float v2f;
typedef __attribute__((ext_vector_type(8))) float v8f;

#define NB     8      // batches
#define CI     128    // input channels
#define CO     128    // output channels
#define NPTS   8192   // spatial points
#define KMJ    32     // KM (j dim)
#define MODES  256    // spectral modes
#define KSEG   2048   // K split for stage A (NPTS / 4)

// ---------------------------------------------------------------------------
// Stage A: x_hat[b,c,m] = sum_n x[b,c,n] * wbases[b,n,m]
// grid = NB * (CI/16) * (NPTS/KSEG) = 8*8*4 = 256 blocks, 256 threads (8 waves)
// Each block: one batch, one 16-row c-tile, one K segment; waves cover 2
// mode-tiles each (8 waves * 32 modes = 256). Partial sums atomically added.
// ---------------------------------------------------------------------------
__global__ __launch_bounds__(256) void stageA(const float* __restrict__ x,
                                              const float* __restrict__ wb,
                                              float* __restrict__ xhat) {
    const int nseg = NPTS / KSEG;
    const int b    = blockIdx.x / ((CI / 16) * nseg);
    const int rest = blockIdx.x % ((CI / 16) * nseg);
    const int ct   = rest / nseg;
    const int ks   = rest % nseg;
    const int c0   = ct * 16;
    const int kbeg = ks * KSEG;

    const int t    = threadIdx.x;
    const int wave = t >> 5;
    const int lane = t & 31;
    const int lm   = lane & 15;   // row/col within tile
    const int lh   = lane >> 4;   // lane half selects K pair

    __shared__ float xs[16][32 + 2];        // x chunk, padded stride
    __shared__ float ws[32][256 + 4];       // wbases chunk, padded stride

    const float* xb  = x  + ((size_t)b * CI + c0) * NPTS;
    const float* wbb = wb + (size_t)b * NPTS * MODES;

    v8f acc0 = {};
    v8f acc1 = {};
    const int m0 = wave * 32;               // this wave: modes [m0, m0+32)

    for (int k0 = kbeg; k0 < kbeg + KSEG; k0 += 32) {
        __syncthreads();
        // --- stage x chunk: 16 rows x 32 K = 512 floats, 2 per thread ---
        {
            const int idx = t * 2;
            const int r = idx >> 5, c = idx & 31;
            const float2 v = *(const float2*)(xb + (size_t)r * NPTS + k0 + c);
            xs[r][c]     = v.x;
            xs[r][c + 1] = v.y;
        }
        // --- stage wbases chunk: 32 K-rows x 256 modes = 2048 float4 ---
        #pragma unroll
        for (int i = 0; i < 8; ++i) {
            const int idx = (i * 256 + t) * 4;
            const int r = idx >> 8, c = idx & 255;
            const float4 v = *(const float4*)(wbb + (size_t)(k0 + r) * MODES + c);
            ws[r][c]     = v.x;
            ws[r][c + 1] = v.y;
            ws[r][c + 2] = v.z;
            ws[r][c + 3] = v.w;
        }
        __syncthreads();

        #pragma unroll
        for (int kk = 0; kk < 32; kk += 4) {
            v2f a;
            a.x = xs[lm][kk + lh * 2];
            a.y = xs[lm][kk + lh * 2 + 1];
            v2f b0, b1;
            b0.x = ws[kk + lh * 2][m0 + lm];
            b0.y = ws[kk + lh * 2 + 1][m0 + lm];
            b1.x = ws[kk + lh * 2][m0 + 16 + lm];
            b1.y = ws[kk + lh * 2 + 1][m0 + 16 + lm];
            acc0 = __builtin_amdgcn_wmma_f32_16x16x4_f32(false, a, false, b0,
                                                         (short)0, acc0, false, false);
            acc1 = __builtin_amdgcn_wmma_f32_16x16x4_f32(false, a, false, b1,
                                                         (short)0, acc1, false, false);
        }
    }

    // C layout: VGPR r -> row (r + lh*8), col lm
    float* outp = xhat + ((size_t)b * CI + c0) * MODES;
    #pragma unroll
    for (int r = 0; r < 8; ++r) {
        atomicAdd(&outp[(size_t)(r + lh * 8) * MODES + m0 + lm],      acc0[r]);
        atomicAdd(&outp[(size_t)(r + lh * 8) * MODES + m0 + 16 + lm], acc1[r]);
    }
}

// ---------------------------------------------------------------------------
// Stage B: y_hat[b,o,k] = sum_i x_hat[b,i,k] * (sum_j weights[i,o,j]*D[j,k])
// grid = MODES (one block per mode k), 256 threads. M_k built in 16-row
// chunks in LDS; each thread owns 4 (b,o) accumulators. ~0.34 GFLOP total.
// ---------------------------------------------------------------------------
__global__ __launch_bounds__(256) void stageB(const float* __restrict__ xhat,
                                              const float* __restrict__ w,
                                              const float* __restrict__ D,
                                              float* __restrict__ yhat) {
    const int k = blockIdx.x;
    const int t = threadIdx.x;

    __shared__ float dcol[KMJ];
    __shared__ float xsl[NB][CI];          // x_hat[:, :, k]
    __shared__ float mk[16][CO + 1];       // chunk of M_k

    if (t < KMJ) dcol[t] = D[(size_t)t * MODES + k];
    for (int i = t; i < NB * CI; i += 256) {
        const int bb = i / CI, ii = i % CI;
        xsl[bb][ii] = xhat[((size_t)bb * CI + ii) * MODES + k];
    }
    __syncthreads();

    const int o    = t & 127;
    const int bgrp = (t >> 7) * 4;         // this thread: b = bgrp..bgrp+3
    float acc[4] = {0.f, 0.f, 0.f, 0.f};

    for (int i0 = 0; i0 < CI; i0 += 16) {
        // build M_k chunk: 16 x 128 values, 8 per thread, 32-long dots over j
        for (int v = t; v < 16 * CO; v += 256) {
            const int ii = v >> 7, oo = v & 127;
            const float4* wp =
                (const float4*)(w + ((size_t)(i0 + ii) * CO + oo) * KMJ);
            float s = 0.f;
            #pragma unroll
            for (int j4 = 0; j4 < 8; ++j4) {
                const float4 wv = wp[j4];
                s += wv.x * dcol[j4 * 4 + 0];
                s += wv.y * dcol[j4 * 4 + 1];
                s += wv.z * dcol[j4 * 4 + 2];
                s += wv.w * dcol[j4 * 4 + 3];
            }
            mk[ii][oo] = s;
        }
        __syncthreads();
        #pragma unroll
        for (int ii = 0; ii < 16; ++ii) {
            const float m = mk[ii][o];
            acc[0] += xsl[bgrp + 0][i0 + ii] * m;
            acc[1] += xsl[bgrp + 1][i0 + ii] * m;
            acc[2] += xsl[bgrp + 2][i0 + ii] * m;
            acc[3] += xsl[bgrp + 3][i0 + ii] * m;
        }
        __syncthreads();
    }

    #pragma unroll
    for (int q = 0; q < 4; ++q)
        yhat[((size_t)(bgrp + q) * CO + o) * MODES + k] = acc[q];
}

// ---------------------------------------------------------------------------
// Stage C: y[b,c,n] = sum_k y_hat[b,c,k] * bases[b,n,k]
// grid = NB * (CO/16) * (NPTS/128) = 4096 blocks, 256 threads (8 waves).
// y_hat c-tile (16x256) cached in LDS; bases fragments are contiguous
// per-lane float2 loads (modes = fast axis). One 16x16 tile per wave.
// ---------------------------------------------------------------------------
__global__ __launch_bounds__(256) void stageC(const float* __restrict__ yhat,
                                              const float* __restrict__ bases,
                                              float* __restrict__ y) {
    const int ngroups = NPTS / 128;
    const int ng = blockIdx.x % ngroups;
    const int ct = (blockIdx.x / ngroups) % (CO / 16);
    const int b  = blockIdx.x / (ngroups * (CO / 16));
    const int c0 = ct * 16;

    const int t    = threadIdx.x;
    const int wave = t >> 5;
    const int lane = t & 31;
    const int lm   = lane & 15;
    const int lh   = lane >> 4;

    __shared__ float ysl[16][MODES + 2];

    // stage y_hat tile: 16 x 256 = 1024 float4, 4 per thread
    const float* yh = yhat + ((size_t)b * CO + c0) * MODES;
    #pragma unroll
    for (int i = 0; i < 4; ++i) {
        const int idx = (i * 256 + t) * 4;
        const int r = idx >> 8, c = idx & 255;
        const float4 v = *(const float4*)(yh + (size_t)r * MODES + c);
        ysl[r][c]     = v.x;
        ysl[r][c + 1] = v.y;
        ysl[r][c + 2] = v.z;
        ysl[r][c + 3] = v.w;
    }
    __syncthreads();

    const int n0 = ng * 128 + wave * 16;
    // B frag: lane lm -> spatial row (n0+lm), K pair at (kk + lh*2): contiguous
    const float* bp = bases + ((size_t)b * NPTS + n0 + lm) * MODES + lh * 2;

    v8f acc = {};
    #pragma unroll 8
    for (int kk = 0; kk < MODES; kk += 4) {
        v2f a, bb;
        a.x = ysl[lm][kk + lh * 2];
        a.y = ysl[lm][kk + lh * 2 + 1];
        const float2 bv = *(const float2*)(bp + kk);
        bb.x = bv.x;
        bb.y = bv.y;
        acc = __builtin_amdgcn_wmma_f32_16x16x4_f32(false, a, false, bb,
                                                    (short)0, acc, false, false);
    }

    float* yo = y + ((size_t)b * CO + c0) * NPTS + n0;
    #pragma unroll
    for (int r = 0; r < 8; ++r)
        yo[(size_t)(r + lh * 8) * NPTS + lm] = acc[r];
}

// ---------------------------------------------------------------------------
extern "C" void kernel_launch(void* const* d_in, const int* in_sizes, int n_in,
                              void* d_out, int out_size, void* d_ws, size_t ws_size,
                              hipStream_t stream) {
    const float* x       = (const float*)d_in[0];  // (8,128,8192)
    const float* wbases  = (const float*)d_in[1];  // (8,8192,256)
    const float* bases   = (const float*)d_in[2];  // (8,8192,256)
    const float* weights = (const float*)d_in[3];  // (128,128,32)
    const float* D       = (const float*)d_in[4];  // (32,256)
    float*       y       = (float*)d_out;          // (8,128,8192)

    const size_t xhat_elems = (size_t)NB * CI * MODES;      // 262144
    float* xhat = (float*)d_ws;
    float* yhat = (float*)d_ws + xhat_elems;

    // stage A accumulates with atomics -> zero x_hat each call (graph-safe)
    hipMemsetAsync(xhat, 0, xhat_elems * sizeof(float), stream);

    stageA<<<NB * (CI / 16) * (NPTS / KSEG), 256, 0, stream>>>(x, wbases, xhat);
    stageB<<<MODES, 256, 0, stream>>>(xhat, weights, D, yhat);
    stageC<<<NB * (CO / 16) * (NPTS / 128), 256, 0, stream>>>(yhat, bases, y);
}